// TextKernelLoss_60421599920384
// MI455X (gfx1250) — compile-verified
//
#include <hip/hip_runtime.h>

// ---------------------------------------------------------------------------
// PSENet text/kernel loss for MI455X (gfx1250, wave32).
// Memory-bound: ~210MB traffic -> ~9us at 23.3 TB/s HBM; pass 2's 79MB
// working set is L2-resident (192MB L2). All plane reads are float4
// (GLOBAL_LOAD_B128) for maximal bytes/VMEM-instruction. OHEM order statistic
// via 1024-bin histogram of sigmoid scores (1/1024 value granularity).
// Cross-lane partial-sum reductions run on the matrix pipe with
// V_WMMA_F32_16X16X4_F32 (B = ones => row sums), EXEC all-1s at call sites.
// ---------------------------------------------------------------------------

#define TPB     256
#define NBATCH  16
#define HH      640
#define WW      640
#define HWTOT   (HH * WW)           // 409600 floats per plane
#define NBINS   1024
#define VEC     4
#define BPI     100                 // blocks per image
#define CHUNK   (HWTOT / BPI)       // 4096 floats per block
#define CHUNKV  (CHUNK / VEC)       // 1024 float4 per block
#define ITERS   (CHUNKV / TPB)      // 4 (exact tiling -> EXEC all-1s)
#define OHEMR   3
#define EPSF    1e-6f

typedef float v2f __attribute__((ext_vector_type(2)));
typedef float v8f __attribute__((ext_vector_type(8)));

__device__ __forceinline__ float lane_bcast_f(float x, int lane) {
  return __int_as_float(__builtin_amdgcn_readlane(__float_as_int(x), lane));
}

// Sum v over the 32 lanes of this wave using the matrix pipe.
// A(16x4): lane L<16 holds A[L][0]=v; lane L>=16 holds A[L-16][2]=v; rest 0.
// B(4x16) = ones.  D[m][n] = v[m] + v[m+16] for every n.  Per-lane sum of the
// 8 D VGPRs leaves half-wave sums in lanes 0-15 / 16-31; two readlanes finish.
__device__ __forceinline__ float wave_sum_wmma(float v) {
  v2f a; a.x = v;    a.y = 0.0f;
  v2f b; b.x = 1.0f; b.y = 1.0f;
  v8f c = {};
  c = __builtin_amdgcn_wmma_f32_16x16x4_f32(false, a, false, b,
                                            (short)0, c, false, false);
  float s = ((c[0] + c[1]) + (c[2] + c[3])) + ((c[4] + c[5]) + (c[6] + c[7]));
  return lane_bcast_f(s, 0) + lane_bcast_f(s, 31);
}

// ---------------------------------------------------------------------------
// Pass 1: per-image counts (pos, neg), histogram of negative sigmoid scores,
// and kernel-dice partial sums (threshold-independent).
// gacc per image: [0]=pos [1]=neg [2]=PGk [3]=P2k [4]=G2k [5]=PGt [6]=P2t [7]=G2t
// ---------------------------------------------------------------------------
__global__ __launch_bounds__(TPB) void pass1_kernel(
    const float* __restrict__ preds, const float* __restrict__ targets,
    const float* __restrict__ eff, unsigned int* __restrict__ ghist,
    float* __restrict__ gacc) {
  __shared__ unsigned int lhist[NBINS];
  __shared__ float lred[5][8];

  const int n     = blockIdx.x / BPI;
  const int chunk = blockIdx.x % BPI;
  const int base  = chunk * CHUNKV + (int)threadIdx.x;

  for (int i = threadIdx.x; i < NBINS; i += TPB) lhist[i] = 0u;
  __syncthreads();

  const float4* __restrict__ p0v =
      (const float4*)(preds   + (size_t)(n * 6 + 0) * HWTOT);
  const float4* __restrict__ p1v =
      (const float4*)(preds   + (size_t)(n * 6 + 1) * HWTOT);
  const float4* __restrict__ t0v =
      (const float4*)(targets + (size_t)(n * 2 + 0) * HWTOT);
  const float4* __restrict__ t1v =
      (const float4*)(targets + (size_t)(n * 2 + 1) * HWTOT);
  const float4* __restrict__ epv =
      (const float4*)(eff     + (size_t)n * HWTOT);

  float pos = 0.0f, negc = 0.0f, PGk = 0.0f, P2k = 0.0f, G2k = 0.0f;
#pragma unroll
  for (int it = 0; it < ITERS; ++it) {
    const int i = base + it * TPB;
    const float4 x0 = p0v[i];
    const float4 x1 = p1v[i];
    const float4 t0 = t0v[i];
    const float4 t1 = t1v[i];
    const float4 ev = epv[i];
    const float xs0[VEC] = {x0.x, x0.y, x0.z, x0.w};
    const float xs1[VEC] = {x1.x, x1.y, x1.z, x1.w};
    const float ts0[VEC] = {t0.x, t0.y, t0.z, t0.w};
    const float ts1[VEC] = {t1.x, t1.y, t1.z, t1.w};
    const float es [VEC] = {ev.x, ev.y, ev.z, ev.w};
#pragma unroll
    for (int c = 0; c < VEC; ++c) {
      const float pt = 1.0f / (1.0f + __expf(-xs0[c]));  // sigmoid(pred_text)
      const float pk = 1.0f / (1.0f + __expf(-xs1[c]));  // sigmoid(pred_kernel)
      const bool tpos = (ts0[c] > 0.5f);
      const bool eon  = (es[c]  > 0.5f);
      pos += (tpos && eon) ? 1.0f : 0.0f;
      if (!tpos) {
        negc += 1.0f;
        int bbin = (int)(pt * (float)NBINS);
        bbin = bbin < 0 ? 0 : (bbin > NBINS - 1 ? NBINS - 1 : bbin);
        atomicAdd(&lhist[bbin], 1u);
      }
      // kernel dice: sel_k = (p_text > 0.5) & (E > 0.5)
      const float mk = (pt > 0.5f && eon) ? 1.0f : 0.0f;
      const float P  = pk * mk;
      const float T  = (ts1[c] > 0.5f ? 1.0f : 0.0f) * mk;
      PGk += P * T;
      P2k += P * P;
      G2k += T * T;
    }
  }

  // Converged here: WMMA wave reductions (EXEC all-1s), LDS, 1 atomic each.
  const float w0 = wave_sum_wmma(pos);
  const float w1 = wave_sum_wmma(negc);
  const float w2 = wave_sum_wmma(PGk);
  const float w3 = wave_sum_wmma(P2k);
  const float w4 = wave_sum_wmma(G2k);
  const int wave = threadIdx.x >> 5;
  const int lane = threadIdx.x & 31;
  if (lane == 0) {
    lred[0][wave] = w0; lred[1][wave] = w1; lred[2][wave] = w2;
    lred[3][wave] = w3; lred[4][wave] = w4;
  }
  __syncthreads();
  if (threadIdx.x < 5) {
    float t = 0.0f;
#pragma unroll
    for (int w = 0; w < 8; ++w) t += lred[threadIdx.x][w];
    atomicAdd(&gacc[n * 8 + (int)threadIdx.x], t);
  }
  for (int i = threadIdx.x; i < NBINS; i += TPB) {
    const unsigned int cnt = lhist[i];
    if (cnt) atomicAdd(&ghist[n * NBINS + i], cnt);
  }
}

// ---------------------------------------------------------------------------
// Threshold selection: descending cumulative scan of the per-image histogram.
// ---------------------------------------------------------------------------
__global__ void select_kernel(const unsigned int* __restrict__ ghist,
                              const float* __restrict__ gacc,
                              float* __restrict__ thr,
                              float* __restrict__ fb) {
  const int n = blockIdx.x;
  if (threadIdx.x == 0) {
    const long long pos  = (long long)(gacc[n * 8 + 0] + 0.5f);
    const long long negc = (long long)(gacc[n * 8 + 1] + 0.5f);
    long long neg_num = pos * OHEMR;
    if (neg_num > negc) neg_num = negc;
    if (pos == 0 || neg_num == 0) {
      thr[n] = 0.0f;
      fb[n]  = 1.0f;   // fallback: sel = E (raw float weights)
      return;
    }
    const unsigned int* __restrict__ h = ghist + n * NBINS;
    long long cum = 0;
    int bsel = 0;
    for (int b = NBINS - 1; b >= 0; --b) {
      cum += (long long)h[b];
      if (cum >= neg_num) { bsel = b; break; }
    }
    thr[n] = (float)bsel / (float)NBINS;  // lower edge of selected bin
    fb[n]  = 0.0f;
  }
}

// ---------------------------------------------------------------------------
// Pass 2: text-dice partial sums with the OHEM mask (L2-resident re-read).
// ---------------------------------------------------------------------------
__global__ __launch_bounds__(TPB) void pass2_kernel(
    const float* __restrict__ preds, const float* __restrict__ targets,
    const float* __restrict__ eff, const float* __restrict__ thr,
    const float* __restrict__ fb, float* __restrict__ gacc) {
  __shared__ float lred[3][8];

  const int n     = blockIdx.x / BPI;
  const int chunk = blockIdx.x % BPI;
  const int base  = chunk * CHUNKV + (int)threadIdx.x;

  const float4* __restrict__ p0v =
      (const float4*)(preds   + (size_t)(n * 6) * HWTOT);
  const float4* __restrict__ t0v =
      (const float4*)(targets + (size_t)(n * 2) * HWTOT);
  const float4* __restrict__ epv =
      (const float4*)(eff     + (size_t)n * HWTOT);
  const float th  = thr[n];   // uniform -> scalar loads
  const float fbn = fb[n];

  float PG = 0.0f, P2 = 0.0f, G2 = 0.0f;
#pragma unroll
  for (int it = 0; it < ITERS; ++it) {
    const int i = base + it * TPB;
    const float4 x0 = p0v[i];
    const float4 t0 = t0v[i];
    const float4 ev = epv[i];
    const float xs0[VEC] = {x0.x, x0.y, x0.z, x0.w};
    const float ts0[VEC] = {t0.x, t0.y, t0.z, t0.w};
    const float es [VEC] = {ev.x, ev.y, ev.z, ev.w};
#pragma unroll
    for (int c = 0; c < VEC; ++c) {
      const float pt = 1.0f / (1.0f + __expf(-xs0[c]));
      float m;
      if (fbn > 0.5f) {  // uniform branch per image
        m = es[c];
      } else {
        m = (((pt >= th) || (ts0[c] > 0.5f)) && (es[c] > 0.5f)) ? 1.0f : 0.0f;
      }
      const float P = pt * m;
      const float T = (ts0[c] > 0.5f ? 1.0f : 0.0f) * m;
      PG += P * T;
      P2 += P * P;
      G2 += T * T;
    }
  }

  const float w0 = wave_sum_wmma(PG);
  const float w1 = wave_sum_wmma(P2);
  const float w2 = wave_sum_wmma(G2);
  const int wave = threadIdx.x >> 5;
  const int lane = threadIdx.x & 31;
  if (lane == 0) { lred[0][wave] = w0; lred[1][wave] = w1; lred[2][wave] = w2; }
  __syncthreads();
  if (threadIdx.x < 3) {
    float t = 0.0f;
#pragma unroll
    for (int w = 0; w < 8; ++w) t += lred[threadIdx.x][w];
    atomicAdd(&gacc[n * 8 + 5 + (int)threadIdx.x], t);
  }
}

// ---------------------------------------------------------------------------
// Finalize: 16 text losses then 16 kernel losses.
// ---------------------------------------------------------------------------
__global__ void finalize_kernel(const float* __restrict__ gacc,
                                float* __restrict__ out) {
  const int n = threadIdx.x;
  if (n < NBATCH) {
    const float PGt = gacc[n * 8 + 5];
    const float P2t = gacc[n * 8 + 6];
    const float G2t = gacc[n * 8 + 7];
    out[n] = 1.0f - 2.0f * PGt / (P2t + G2t + EPSF);
    const float PGk = gacc[n * 8 + 2];
    const float P2k = gacc[n * 8 + 3];
    const float G2k = gacc[n * 8 + 4];
    out[NBATCH + n] = 1.0f - 2.0f * PGk / (P2k + G2k + EPSF);
  }
}

extern "C" void kernel_launch(void* const* d_in, const int* in_sizes, int n_in,
                              void* d_out, int out_size, void* d_ws, size_t ws_size,
                              hipStream_t stream) {
  (void)in_sizes; (void)n_in; (void)out_size; (void)ws_size;
  const float* preds   = (const float*)d_in[0];   // [16,6,640,640] f32
  const float* targets = (const float*)d_in[1];   // [16,2,640,640] f32
  const float* eff     = (const float*)d_in[2];   // [16,640,640]   f32
  float* out = (float*)d_out;                     // [32] f32 (text, kernel)

  // Workspace: hist[16][1024] u32 | gacc[16][8] f32 | thr[16] f32 | fb[16] f32
  unsigned int* ghist = (unsigned int*)d_ws;
  float* gacc = (float*)((char*)d_ws + (size_t)NBATCH * NBINS * sizeof(unsigned int));
  float* thr  = gacc + NBATCH * 8;
  float* fb   = thr + NBATCH;
  const size_t zbytes = (size_t)NBATCH * NBINS * sizeof(unsigned int)
                      + (size_t)(NBATCH * 8 + NBATCH + NBATCH) * sizeof(float);
  hipMemsetAsync(d_ws, 0, zbytes, stream);

  dim3 grid(NBATCH * BPI), block(TPB);
  pass1_kernel<<<grid, block, 0, stream>>>(preds, targets, eff, ghist, gacc);
  select_kernel<<<NBATCH, 32, 0, stream>>>(ghist, gacc, thr, fb);
  pass2_kernel<<<grid, block, 0, stream>>>(preds, targets, eff, thr, fb, gacc);
  finalize_kernel<<<1, 32, 0, stream>>>(gacc, out);
}